// attention_compute_28063316312630
// MI455X (gfx1250) — compile-verified
//
#include <hip/hip_runtime.h>

#define B_ 64
#define S_ 64
#define F_ 512
#define E_ 1024
#define D_ 1024

typedef __attribute__((ext_vector_type(2))) float v2f;
typedef __attribute__((ext_vector_type(8))) float v8f;

__global__ __launch_bounds__(256) void fused_feat_attn_kernel(
    const float* __restrict__ features,          // (B, F, D) f32
    const long long* __restrict__ feature_lens,  // (B,) i64
    const float* __restrict__ W,                 // (E + D,) f32
    float* __restrict__ out)                     // (B, S, D) f32
{
    __shared__ float sWf[D_];     // W_f staged in LDS (4 KB)
    __shared__ float sP[F_];      // scores -> softmax probabilities (2 KB)
    __shared__ float sRed[256];   // block reduction scratch (1 KB)

    const int b    = blockIdx.x;
    const int tid  = threadIdx.x;
    const int lane = tid & 31;
    const int wave = tid >> 5;

    const float* fb = features + (size_t)b * F_ * D_;
    const long long len = feature_lens[b];

    // ---- stage W_f into LDS ----
    for (int i = tid; i < D_; i += 256) sWf[i] = W[E_ + i];
    __syncthreads();

    // ---- phase 1: s_ftr[f] = features[b,f,:] . W_f   (wave-per-row) ----
    for (int f = wave; f < F_; f += 8) {
        const float* row = fb + (size_t)f * D_;
        float acc = 0.0f;
        for (int d = lane; d < D_; d += 32)
            acc = fmaf(row[d], sWf[d], acc);
        #pragma unroll
        for (int off = 16; off > 0; off >>= 1)
            acc += __shfl_xor(acc, off, 32);
        if (lane == 0) sP[f] = acc;
    }
    __syncthreads();

    // ---- phase 2: masked softmax over f (shift-invariance removed s_emb & bias) ----
    float lmax = -3.402823466e+38f;
    for (int f = tid; f < F_; f += 256)
        if (f < len) lmax = fmaxf(lmax, sP[f]);
    sRed[tid] = lmax;
    __syncthreads();
    #pragma unroll
    for (int s = 128; s > 0; s >>= 1) {
        if (tid < s) sRed[tid] = fmaxf(sRed[tid], sRed[tid + s]);
        __syncthreads();
    }
    const float gmax = sRed[0];
    __syncthreads();

    float lsum = 0.0f;
    for (int f = tid; f < F_; f += 256) {
        const float e = (f < len) ? __expf(sP[f] - gmax) : 0.0f;
        sP[f] = e;
        lsum += e;
    }
    sRed[tid] = lsum;
    __syncthreads();
    #pragma unroll
    for (int s = 128; s > 0; s >>= 1) {
        if (tid < s) sRed[tid] += sRed[tid + s];
        __syncthreads();
    }
    const float inv = 1.0f / sRed[0];
    for (int f = tid; f < F_; f += 256) sP[f] *= inv;
    __syncthreads();

    // ---- phase 3: out tile = (16 dup rows of p) x features via V_WMMA_F32_16X16X4_F32 ----
    const int h    = lane >> 4;   // K half selector (0: K=0,1  1: K=2,3)
    const int ncol = lane & 15;   // N column within tile
    float* ob = out + (size_t)b * S_ * D_;

    for (int t = wave; t < D_ / 16; t += 8) {
        const int d0 = t * 16;
        v8f acc = {};
        for (int kk = 0; kk < F_; kk += 4) {
            const int ka = kk + 2 * h;
            v2f a, bm;
            a.x = sP[ka];
            a.y = sP[ka + 1];
            const float* p0 = fb + (size_t)ka * D_ + d0 + ncol;
            bm.x = p0[0];
            bm.y = p0[D_];
            acc = __builtin_amdgcn_wmma_f32_16x16x4_f32(
                false, a, false, bm, (short)0, acc, false, false);
        }
        // C/D layout: VGPR r -> row (8*h + r), col ncol. All 16 rows are the
        // broadcast output vector; replicate the tile to s = 0,16,32,48.
        const int mbase = h * 8;
        #pragma unroll
        for (int r = 0; r < 8; ++r) {
            const float v = acc[r];
            const int m = mbase + r;
            #pragma unroll
            for (int s0 = 0; s0 < S_; s0 += 16)
                ob[(size_t)(s0 + m) * D_ + d0 + ncol] = v;
        }
    }
}

extern "C" void kernel_launch(void* const* d_in, const int* in_sizes, int n_in,
                              void* d_out, int out_size, void* d_ws, size_t ws_size,
                              hipStream_t stream) {
    // inputs: 0=sen_emb (unused: cancels in softmax), 1=features,
    //         2=feature_lens (i64), 3=W, 4=b (unused: cancels in softmax)
    const float*     features = (const float*)d_in[1];
    const long long* flens    = (const long long*)d_in[2];
    const float*     W        = (const float*)d_in[3];
    float*           out      = (float*)d_out;
    (void)in_sizes; (void)n_in; (void)out_size; (void)d_ws; (void)ws_size;

    fused_feat_attn_kernel<<<dim3(B_), dim3(256), 0, stream>>>(features, flens, W, out);
}